// MamboPoolingWithLineGraph_31086973288656
// MI455X (gfx1250) — compile-verified
//
#include <hip/hip_runtime.h>
#include <hip/hip_bf16.h>

typedef __attribute__((ext_vector_type(2))) float v2f;
typedef __attribute__((ext_vector_type(8))) float v8f;

#define CDIM 128
#define EPB  64    // edges per block (4 row tiles of 16)
#define KC   96    // K chunk for GEMM1 staging (384 = 4 * 96)
#define SMK  100   // padded LDS stride for merged K-chunk (100*lh % 64 -> 16 distinct banks)
#define SL   132   // padded LDS stride for line_x rows

// ---------------------------------------------------------------------------
// Degree kernels
// ---------------------------------------------------------------------------
__global__ __launch_bounds__(256) void k_init_deg(float* __restrict__ deg, int E) {
    int i = blockIdx.x * 256 + threadIdx.x;
    if (i < E) deg[i] = 1.0f;  // self loop
}

__global__ __launch_bounds__(256) void k_deg_atomic(float* __restrict__ deg,
                                                    const int* __restrict__ ldst, int M) {
    int i = blockIdx.x * 256 + threadIdx.x;
    if (i < M) {
        __hip_atomic_fetch_add(&deg[ldst[i]], 1.0f,
                               __ATOMIC_RELAXED, __HIP_MEMORY_SCOPE_AGENT);
    }
}

__global__ __launch_bounds__(256) void k_deg_rsqrt(float* __restrict__ deg, int E) {
    int i = blockIdx.x * 256 + threadIdx.x;
    if (i < E) deg[i] = 1.0f / sqrtf(deg[i]);   // deg >= 1 always (self loop)
}

// ---------------------------------------------------------------------------
// Fused edge-MLP + GCN-weight GEMM:
//   xw[e] = (concat(x[src],x[dst],ea) @ We + be) @ Wg
// Block: 256 threads = 8 waves; wave w owns output columns [16w, 16w+16).
// Block covers 64 edges = 4 row tiles; each B fragment is reused by 4 WMMAs.
// ---------------------------------------------------------------------------
__global__ __launch_bounds__(256) void k_edge_mlp_gemm(
    const float* __restrict__ x, const float* __restrict__ ea,
    const float* __restrict__ We, const float* __restrict__ be,
    const float* __restrict__ Wg,
    const int* __restrict__ src, const int* __restrict__ dst,
    float* __restrict__ xw, int E)
{
    __shared__ __align__(16) float smA[EPB * SMK];   // merged K-chunk [64 x 96]
    __shared__ __align__(16) float lx [EPB * SL];    // line_x rows   [64 x 128]
    __shared__ int s_src[EPB];
    __shared__ int s_dst[EPB];

    const int tid = threadIdx.x;
    const int e0  = blockIdx.x * EPB;

    if (tid < EPB) {
        int e = e0 + tid;
        if (e >= E) e = E - 1;               // duplicate tail rows (stores guarded)
        s_src[tid] = src[e];
        s_dst[tid] = dst[e];
    }
    __syncthreads();

    const int wave  = tid >> 5;              // 0..7 -> column tile
    const int lane  = tid & 31;
    const int lh    = lane & 15;
    const int hi    = lane >> 4;             // half-wave select
    const int khalf = hi * 2;                // K offset of this half in A/B frags
    const int cn    = wave * 16;             // this wave's output column base

    // ---- GEMM1: merged[64x384] @ We[384x128], K streamed in 4 chunks ----
    v8f acc[4] = {};
    for (int kc = 0; kc < 4; ++kc) {
        const int kbase = kc * KC;

        // stage merged[:, kbase : kbase+96] into LDS (coalesced within rows)
        for (int i = tid; i < EPB * KC; i += 256) {
            int row = i / KC;
            int cl  = i - row * KC;
            int c   = kbase + cl;
            float v;
            if (c < 128)      v = x[(long)s_src[row] * CDIM + c];
            else if (c < 256) v = x[(long)s_dst[row] * CDIM + (c - 128)];
            else {
                int e = e0 + row;
                if (e >= E) e = E - 1;
                v = ea[(long)e * CDIM + (c - 256)];
            }
            smA[row * SMK + cl] = v;
        }
        __syncthreads();

        for (int k = 0; k < KC; k += 4) {
            const int kk = k + khalf;
            v2f b;   // B 4x16 fp32 frag: N = cn+lh, K split across half-waves
            b.x = We[(long)(kbase + kk) * CDIM + cn + lh];
            b.y = We[(long)(kbase + kk + 1) * CDIM + cn + lh];
#pragma unroll
            for (int rt = 0; rt < 4; ++rt) {
                v2f a;   // A 16x4 fp32 frag: M = lane&15 within row tile rt
                a.x = smA[(rt * 16 + lh) * SMK + kk];
                a.y = smA[(rt * 16 + lh) * SMK + kk + 1];
                acc[rt] = __builtin_amdgcn_wmma_f32_16x16x4_f32(
                    false, a, false, b, (short)0, acc[rt], false, false);
            }
        }
        __syncthreads();   // before restaging smA
    }

    // ---- bias + keep line_x resident in LDS ----
    const float bb = be[cn + lh];
#pragma unroll
    for (int rt = 0; rt < 4; ++rt) {
#pragma unroll
        for (int r = 0; r < 8; ++r) {
            const int m = rt * 16 + r + (hi << 3);   // C/D: lanes>=16 hold M=r+8
            lx[m * SL + cn + lh] = acc[rt][r] + bb;
        }
    }
    __syncthreads();

    // ---- GEMM2: line_x[64x128] @ Wg[128x128] -> xw (workspace) ----
    v8f acc2[4] = {};
    for (int k = 0; k < CDIM; k += 4) {
        const int kk = k + khalf;
        v2f b;
        b.x = Wg[(long)kk * CDIM + cn + lh];
        b.y = Wg[(long)(kk + 1) * CDIM + cn + lh];
#pragma unroll
        for (int rt = 0; rt < 4; ++rt) {
            v2f a;
            a.x = lx[(rt * 16 + lh) * SL + kk];
            a.y = lx[(rt * 16 + lh) * SL + kk + 1];
            acc2[rt] = __builtin_amdgcn_wmma_f32_16x16x4_f32(
                false, a, false, b, (short)0, acc2[rt], false, false);
        }
    }
#pragma unroll
    for (int rt = 0; rt < 4; ++rt) {
#pragma unroll
        for (int r = 0; r < 8; ++r) {
            const int m = rt * 16 + r + (hi << 3);
            const int e = e0 + m;
            if (e < E) xw[(long)e * CDIM + cn + lh] = acc2[rt][r];
        }
    }
}

// ---------------------------------------------------------------------------
// out[e] = b_gcn + dinv[e]^2 * xw[e]   (self-loop term + bias), float4
// ---------------------------------------------------------------------------
__global__ __launch_bounds__(256) void k_out_init(
    float* __restrict__ out, const float* __restrict__ xw,
    const float* __restrict__ dinv, const float* __restrict__ bg, int E)
{
    int i = blockIdx.x * 256 + threadIdx.x;       // one float4 per thread
    int n4 = E * (CDIM / 4);
    if (i >= n4) return;
    int e  = i >> 5;
    int c0 = (i & 31) * 4;
    float di = dinv[e];
    float s  = di * di;
    const float4 v = ((const float4*)xw)[i];
    float4 o;
    o.x = bg[c0 + 0] + s * v.x;
    o.y = bg[c0 + 1] + s * v.y;
    o.z = bg[c0 + 2] + s * v.z;
    o.w = bg[c0 + 3] + s * v.w;
    ((float4*)out)[i] = o;
}

// ---------------------------------------------------------------------------
// Scatter over line edges: out[col] += dinv[row]*dinv[col] * xw[row]
// 2 line edges per 256-thread block, one column per thread.
// ---------------------------------------------------------------------------
__global__ __launch_bounds__(256) void k_scatter(
    float* __restrict__ out, const float* __restrict__ xw,
    const float* __restrict__ dinv,
    const int* __restrict__ lsrc, const int* __restrict__ ldst, int M)
{
    int m = blockIdx.x * 2 + (threadIdx.x >> 7);
    if (m >= M) return;
    int c   = threadIdx.x & 127;
    int row = lsrc[m];
    int col = ldst[m];
    float s = dinv[row] * dinv[col];
    float v = s * xw[(long)row * CDIM + c];
    __hip_atomic_fetch_add(&out[(long)col * CDIM + c], v,
                           __ATOMIC_RELAXED, __HIP_MEMORY_SCOPE_AGENT);
}

// ---------------------------------------------------------------------------
// leaky_relu epilogue (slope 0.01), float4
// ---------------------------------------------------------------------------
__global__ __launch_bounds__(256) void k_leaky(float* __restrict__ out, int E) {
    int i = blockIdx.x * 256 + threadIdx.x;
    int n4 = E * (CDIM / 4);
    if (i >= n4) return;
    float4 v = ((float4*)out)[i];
    v.x = v.x > 0.0f ? v.x : 0.01f * v.x;
    v.y = v.y > 0.0f ? v.y : 0.01f * v.y;
    v.z = v.z > 0.0f ? v.z : 0.01f * v.z;
    v.w = v.w > 0.0f ? v.w : 0.01f * v.w;
    ((float4*)out)[i] = v;
}

// ---------------------------------------------------------------------------
extern "C" void kernel_launch(void* const* d_in, const int* in_sizes, int n_in,
                              void* d_out, int out_size, void* d_ws, size_t ws_size,
                              hipStream_t stream) {
    const float* x   = (const float*)d_in[0];
    const float* ea  = (const float*)d_in[1];
    const float* We  = (const float*)d_in[2];
    const float* be  = (const float*)d_in[3];
    const float* Wg  = (const float*)d_in[4];
    const float* bg  = (const float*)d_in[5];
    const int*   ei  = (const int*)d_in[6];
    const int*   lei = (const int*)d_in[8];

    const int E = in_sizes[6] / 2;
    const int M = in_sizes[8] / 2;

    const int* src  = ei;
    const int* dst  = ei + E;
    const int* lsrc = lei;
    const int* ldst = lei + M;

    float* out = (float*)d_out;
    float* xw  = (float*)d_ws;                    // [E, 128]
    float* dnv = xw + (size_t)E * CDIM;           // [E] degree -> dinv (in place)

    // 1) degree / normalization
    k_init_deg <<<(E + 255) / 256, 256, 0, stream>>>(dnv, E);
    if (M > 0)
        k_deg_atomic<<<(M + 255) / 256, 256, 0, stream>>>(dnv, ldst, M);
    k_deg_rsqrt<<<(E + 255) / 256, 256, 0, stream>>>(dnv, E);

    // 2) fused WMMA GEMMs -> xw
    k_edge_mlp_gemm<<<(E + EPB - 1) / EPB, 256, 0, stream>>>(x, ea, We, be, Wg,
                                                            src, dst, xw, E);

    // 3) bias + self loop, 4) scatter, 5) leaky relu
    const int n4 = E * (CDIM / 4);
    k_out_init<<<(n4 + 255) / 256, 256, 0, stream>>>(out, xw, dnv, bg, E);
    if (M > 0)
        k_scatter<<<(M + 1) / 2, 256, 0, stream>>>(out, xw, dnv, lsrc, ldst, M);
    k_leaky  <<<(n4 + 255) / 256, 256, 0, stream>>>(out, E);
}